// TaskAlignedAssigner_41987600286322
// MI455X (gfx1250) — compile-verified
//
#include <hip/hip_runtime.h>
#include <hip/hip_bf16.h>
#include <math.h>

#define TOPKN 13
#define EPSF 1e-9f
#define INV_PI2_4 0.40528473456935108578f

typedef __attribute__((ext_vector_type(2))) float v2f;
typedef __attribute__((ext_vector_type(8))) float v8f;

// ---------------------------------------------------------------------------
// K1: fused pairwise kernel.
// One wave computes a 16(gt) x 16(anchor) tile of bbox_scores via
// V_WMMA_F32_16X16X4_F32 (one-hot A => bit-exact vs gather), then each lane
// finishes the 8 (g,n) pairs it owns in the C-fragment: CIoU, keypoint OKS,
// masks, align metric. Writes align[b,g,n] and overlaps[b,g,n].
// ---------------------------------------------------------------------------
__global__ void __launch_bounds__(128)
k1_pairwise(const float* __restrict__ pd_scores, const float* __restrict__ pd_bboxes,
            const float* __restrict__ anc, const int* __restrict__ gt_labels,
            const float* __restrict__ gt_bboxes, const float* __restrict__ mask_gt,
            const float* __restrict__ gt_kkpts, const float* __restrict__ pd_kkpts,
            const float* __restrict__ sigma, const float* __restrict__ stride_t,
            const int* __restrict__ reg_max_p,
            float* __restrict__ alignp, float* __restrict__ ovlp,
            int na, int nmax, int nc, int nk)
{
    __shared__ float s_gt[16][4];
    __shared__ float s_mgt[16];
    __shared__ int   s_lbl[16];

    const int b     = blockIdx.z;
    const int gbase = blockIdx.y << 4;
    const int tid   = threadIdx.x;

    if (tid < 16) {
        int g = gbase + tid;
        if (g < nmax) {
            const float* gb = gt_bboxes + ((size_t)b * nmax + g) * 4;
            s_gt[tid][0] = gb[0]; s_gt[tid][1] = gb[1];
            s_gt[tid][2] = gb[2]; s_gt[tid][3] = gb[3];
            s_mgt[tid] = mask_gt[(size_t)b * nmax + g];
            s_lbl[tid] = gt_labels[(size_t)b * nmax + g];
        } else {
            s_gt[tid][0] = 0.f; s_gt[tid][1] = 0.f;
            s_gt[tid][2] = 0.f; s_gt[tid][3] = 0.f;
            s_mgt[tid] = 0.f; s_lbl[tid] = -1;
        }
    }
    __syncthreads();

    const int wave  = tid >> 5;
    const int lane  = tid & 31;
    const int laneN = lane & 15;
    const int hi    = (lane >= 16) ? 1 : 0;
    const int tile  = blockIdx.x * 4 + wave;
    const int n0    = tile << 4;
    if (n0 >= na) return;                 // whole wave exits together (na % 16 == 0)
    const int n = n0 + laneN;

    const float* srow = pd_scores + ((size_t)b * na + n) * nc;
    __builtin_prefetch(srow, 0, 3);       // global_prefetch_b8

    // ---- bbox_scores tile: raw[i] = pd_scores[b, n, lbl[gbase + hi*8 + i]] ----
    float raw[8];
#if __has_builtin(__builtin_amdgcn_wmma_f32_16x16x4_f32)
    {
        // Two independent accumulators break the WMMA D->C RAW chain.  With a
        // one-hot A matrix each output gets exactly one nonzero contribution,
        // so acc0+acc1 is bit-exact vs a single accumulator.
        v8f acc0 = {}, acc1 = {};
        const int myl     = s_lbl[laneN];  // A fragment: M = lane & 15
        const int khi     = hi * 2;        // K split across half-waves (ISA 7.12.2)
        const int ks_main = nc >> 2;       // provably in-bounds 4-wide K steps
        int ks = 0;
        for (; ks + 2 <= ks_main; ks += 2) {
            const int kb0 = (ks << 2) + khi;
            const int kb1 = kb0 + 4;
            v2f a0, a1;
            a0.x = (myl == kb0)     ? 1.f : 0.f;
            a0.y = (myl == kb0 + 1) ? 1.f : 0.f;
            a1.x = (myl == kb1)     ? 1.f : 0.f;
            a1.y = (myl == kb1 + 1) ? 1.f : 0.f;
            const v2f b0 = *(const v2f*)(srow + kb0);   // global_load_b64, aligned
            const v2f b1 = *(const v2f*)(srow + kb1);
            acc0 = __builtin_amdgcn_wmma_f32_16x16x4_f32(false, a0, false, b0,
                                                         (short)0, acc0, false, false);
            acc1 = __builtin_amdgcn_wmma_f32_16x16x4_f32(false, a1, false, b1,
                                                         (short)0, acc1, false, false);
        }
        if (ks < ks_main) {                 // odd leftover full step (uniform branch)
            const int kb0 = (ks << 2) + khi;
            v2f a0;
            a0.x = (myl == kb0)     ? 1.f : 0.f;
            a0.y = (myl == kb0 + 1) ? 1.f : 0.f;
            const v2f b0 = *(const v2f*)(srow + kb0);
            acc0 = __builtin_amdgcn_wmma_f32_16x16x4_f32(false, a0, false, b0,
                                                         (short)0, acc0, false, false);
        }
        if (nc & 3) {                       // ragged tail (uniform branch; dead for nc=80)
            const int kb0 = (ks_main << 2) + khi;
            v2f a0, b0;
            a0.x = (kb0     < nc && myl == kb0)     ? 1.f : 0.f;
            a0.y = (kb0 + 1 < nc && myl == kb0 + 1) ? 1.f : 0.f;
            b0.x = (kb0     < nc) ? srow[kb0]     : 0.f;
            b0.y = (kb0 + 1 < nc) ? srow[kb0 + 1] : 0.f;
            acc0 = __builtin_amdgcn_wmma_f32_16x16x4_f32(false, a0, false, b0,
                                                         (short)0, acc0, false, false);
        }
        const v8f accs = acc0 + acc1;
        for (int i = 0; i < 8; ++i) raw[i] = accs[i];
    }
#else
    for (int i = 0; i < 8; ++i) {
        int l = s_lbl[hi * 8 + i];
        raw[i] = ((unsigned)l < (unsigned)nc) ? srow[l] : 0.f;
    }
#endif

    // ---- per-anchor data ----
    const float4 pb4 = *(const float4*)(pd_bboxes + ((size_t)b * na + n) * 4);
    const float px1 = pb4.x, py1 = pb4.y, px2 = pb4.z, py2 = pb4.w;
    const float ax = anc[2 * n], ay = anc[2 * n + 1];
    const float st = stride_t[n];
    const int   reg_max = *reg_max_p;
    const float aps_max = st * (float)(reg_max - 1) * 2.f;
    const float w2 = px2 - px1;
    const float h2 = py2 - py1 + 1e-7f;
    const float atan_p = atanf(w2 / h2);
    const float* pkb = pd_kkpts + ((size_t)b * na + n) * nk * 3;

    for (int i = 0; i < 8; ++i) {
        const int gl = hi * 8 + i;
        const int g  = gbase + gl;
        if (g >= nmax) continue;

        const float gx1 = s_gt[gl][0], gy1 = s_gt[gl][1];
        const float gx2 = s_gt[gl][2], gy2 = s_gt[gl][3];
        const float mgt = s_mgt[gl];

        const float w1  = gx2 - gx1;
        const float h1r = gy2 - gy1;
        const float h1  = h1r + 1e-7f;
        const float gsz = (w1 + h1r) * 0.5f;

        // mask_in_gts * size-filter * mask_gt
        const float din = fminf(fminf(ax - gx1, ay - gy1), fminf(gx2 - ax, gy2 - ay));
        const float m = ((din > EPSF) && (aps_max - gsz >= EPSF)) ? mgt : 0.f;

        // CIoU(gt, pd), clipped at 0
        const float iw = fmaxf(fminf(gx2, px2) - fmaxf(gx1, px1), 0.f);
        const float ih = fmaxf(fminf(gy2, py2) - fmaxf(gy1, py1), 0.f);
        const float inter = iw * ih;
        const float uni = w1 * h1 + w2 * h2 - inter + 1e-7f;
        const float iou = inter / uni;
        const float cw = fmaxf(gx2, px2) - fminf(gx1, px1);
        const float ch = fmaxf(gy2, py2) - fminf(gy1, py1);
        const float c2 = cw * cw + ch * ch + 1e-7f;
        const float ddx = px1 + px2 - gx1 - gx2;
        const float ddy = py1 + py2 - gy1 - gy2;
        const float rho2 = (ddx * ddx + ddy * ddy) * 0.25f;
        const float dat = atan_p - atanf(w1 / h1);
        const float v = INV_PI2_4 * dat * dat;
        const float aco = v / (v - iou + 1.f + 1e-7f);
        const float iouc = fmaxf(iou - (rho2 / c2 + v * aco), 0.f);

        // keypoint OKS
        const float area = w1 * h1r * 0.53f;
        float kacc = 0.f, ksum = 0.f;
        const float* gkb = gt_kkpts + ((size_t)b * nmax + g) * nk * 3;
        for (int k = 0; k < nk; ++k) {
            const float km = (gkb[k * 3 + 2] != 0.f) ? 1.f : 0.f;
            const float dx = pkb[k * 3 + 0] - gkb[k * 3 + 0];
            const float dy = pkb[k * 3 + 1] - gkb[k * 3 + 1];
            const float d  = dx * dx + dy * dy;
            const float sg = sigma[k];
            const float e  = d / (4.f * sg * sg) / (area + 1e-7f) * 0.5f;
            kacc += __expf(-e) * km;
            ksum += km;
        }
        const float kiou = fmaxf(kacc / (ksum + 1e-7f), 0.f);

        const float o  = (iouc + kiou) * 0.5f * m;
        const float o2 = o * o;
        const float av = raw[i] * m * (o2 * o2 * o2);   // score^1 * overlaps^6

        const size_t base = ((size_t)b * nmax + g) * na + n;
        alignp[base] = av;
        ovlp[base]   = o;
    }
}

// ---------------------------------------------------------------------------
// K2: top-13 per (b,g) row over na anchors (row cached in LDS), then apply
// in-gt + size filters to selected anchors -> candidate indices (or -1).
// ---------------------------------------------------------------------------
__global__ void __launch_bounds__(256)
k2_topk(const float* __restrict__ alignp, const float* __restrict__ gt_bboxes,
        const float* __restrict__ mask_gt, const float* __restrict__ anc,
        const float* __restrict__ stride_t, const int* __restrict__ reg_max_p,
        int* __restrict__ cand, int na, int nmax)
{
    extern __shared__ float srow[];
    __shared__ float rv[256];
    __shared__ int   ri[256];
    __shared__ int   ssel[TOPKN];

    const int row = blockIdx.x;
    const int tid = threadIdx.x;
    const float mgt = mask_gt[row];
    if (!(mgt > 0.f)) {                    // invalid gt row: no candidates
        if (tid < TOPKN) cand[row * TOPKN + tid] = -1;
        return;
    }

    const float* ar = alignp + (size_t)row * na;
    for (int nI = tid; nI < na; nI += 256) srow[nI] = ar[nI];
    __syncthreads();

    for (int k = 0; k < TOPKN; ++k) {
        int lsel[TOPKN];
        for (int j = 0; j < k; ++j) lsel[j] = ssel[j];   // registers, not LDS, in scan
        float best = -INFINITY;
        int   bidx = 0x7fffffff;
        for (int nI = tid; nI < na; nI += 256) {
            bool taken = false;
            for (int j = 0; j < k; ++j) if (lsel[j] == nI) taken = true;
            if (taken) continue;
            const float vv = srow[nI];
            if (vv > best || (vv == best && nI < bidx)) { best = vv; bidx = nI; }
        }
        rv[tid] = best; ri[tid] = bidx;
        __syncthreads();
        for (int s = 128; s > 0; s >>= 1) {
            if (tid < s) {
                if (rv[tid + s] > rv[tid] ||
                    (rv[tid + s] == rv[tid] && ri[tid + s] < ri[tid])) {
                    rv[tid] = rv[tid + s]; ri[tid] = ri[tid + s];
                }
            }
            __syncthreads();
        }
        if (tid == 0) ssel[k] = ri[0];
        __syncthreads();
    }

    if (tid < TOPKN) {
        const int idx = ssel[tid];
        int out = -1;
        if (idx >= 0 && idx < na) {
            const float* gb = gt_bboxes + (size_t)row * 4;
            const float gx1 = gb[0], gy1 = gb[1], gx2 = gb[2], gy2 = gb[3];
            const float axx = anc[2 * idx], ayy = anc[2 * idx + 1];
            const float din = fminf(fminf(axx - gx1, ayy - gy1),
                                    fminf(gx2 - axx, gy2 - ayy));
            const float aps = stride_t[idx] * (float)(*reg_max_p - 1) * 2.f;
            const float gsz = ((gx2 - gx1) + (gy2 - gy1)) * 0.5f;
            if (din > EPSF && (aps - gsz) >= EPSF) out = idx;
        }
        cand[row * TOPKN + tid] = out;
    }
}

// ---------------------------------------------------------------------------
// K3a: scatter candidates -> per-anchor fg count and min gt index.
// ---------------------------------------------------------------------------
__global__ void k3a_scatter(const int* __restrict__ cand, int* __restrict__ fg_cnt,
                            int* __restrict__ tg_min, int na, int nmax, int total)
{
    const int i = blockIdx.x * blockDim.x + threadIdx.x;
    if (i >= total) return;
    const int c = cand[i];
    if (c < 0) return;
    const int row = i / TOPKN;
    const int b = row / nmax, g = row % nmax;
    atomicAdd(&fg_cnt[(size_t)b * na + c], 1);
    atomicMin(&tg_min[(size_t)b * na + c], g);
}

// ---------------------------------------------------------------------------
// K3b: resolve multi-assignment (argmax over overlaps column), pick tgi,
// and accumulate per-gt pos_align / pos_ov maxima (uint-bit atomicMax, vals>=0).
// ---------------------------------------------------------------------------
__global__ void k3b_resolve(const int* __restrict__ fg_cnt, const int* __restrict__ tg_min,
                            const float* __restrict__ ovlp, const float* __restrict__ alignp,
                            int* __restrict__ tgi_buf, unsigned* __restrict__ pa_bits,
                            unsigned* __restrict__ po_bits, int na, int nmax, int total)
{
    const int i = blockIdx.x * blockDim.x + threadIdx.x;
    if (i >= total) return;
    const int b = i / na, n = i % na;
    const int c = fg_cnt[i];
    int tgi = 0;
    if (c == 1) {
        tgi = tg_min[i];
    } else if (c > 1) {
        float best = -1.f; int bg = 0;
        for (int g = 0; g < nmax; ++g) {
            const float vv = ovlp[((size_t)b * nmax + g) * na + n];
            if (vv > best) { best = vv; bg = g; }   // first max, like jnp.argmax
        }
        tgi = bg;
    }
    tgi_buf[i] = tgi;
    if (c > 0) {
        const size_t base = ((size_t)b * nmax + tgi) * na + n;
        atomicMax(&pa_bits[(size_t)b * nmax + tgi], __float_as_uint(alignp[base]));
        atomicMax(&po_bits[(size_t)b * nmax + tgi], __float_as_uint(ovlp[base]));
    }
}

// ---------------------------------------------------------------------------
// K5: finalize per-anchor outputs (labels, boxes, fg, tgi) + norm factor.
// ---------------------------------------------------------------------------
__global__ void k5_final(const int* __restrict__ fg_cnt, const int* __restrict__ tgi_buf,
                         const int* __restrict__ gt_labels, const float* __restrict__ gt_bboxes,
                         const float* __restrict__ alignp,
                         const unsigned* __restrict__ pa_bits, const unsigned* __restrict__ po_bits,
                         float* __restrict__ o_lab, float* __restrict__ o_box,
                         float* __restrict__ o_fg, float* __restrict__ o_tgi,
                         float* __restrict__ norm_buf, int* __restrict__ code_buf,
                         int na, int nmax, int total)
{
    const int i = blockIdx.x * blockDim.x + threadIdx.x;
    if (i >= total) return;
    const int b = i / na, n = i % na;
    const int tgi = tgi_buf[i];
    const bool fg = fg_cnt[i] > 0;
    const int lab = gt_labels[(size_t)b * nmax + tgi];
    const int tlab = lab < 0 ? 0 : lab;
    const float4 gb = *(const float4*)(gt_bboxes + ((size_t)b * nmax + tgi) * 4);
    *(float4*)(o_box + (size_t)i * 4) = gb;
    float norm = 0.f;
    if (fg) {
        const float am = alignp[((size_t)b * nmax + tgi) * na + n];
        const float pa = __uint_as_float(pa_bits[(size_t)b * nmax + tgi]);
        const float po = __uint_as_float(po_bits[(size_t)b * nmax + tgi]);
        norm = am * po / (pa + EPSF);
    }
    o_lab[i] = (float)tlab;
    o_fg[i]  = fg ? 1.f : 0.f;
    o_tgi[i] = (float)tgi;
    norm_buf[i] = norm;
    code_buf[i] = fg ? tlab : -1;
}

// ---------------------------------------------------------------------------
// K6: tscores one-hot scatter * norm.
// ---------------------------------------------------------------------------
__global__ void k6_scores(const float* __restrict__ norm_buf, const int* __restrict__ code_buf,
                          float* __restrict__ o_sco, int nc, long total)
{
    const long i = (long)blockIdx.x * blockDim.x + threadIdx.x;
    if (i >= total) return;
    const long bnn = i / nc;
    const int  cc  = (int)(i % nc);
    o_sco[i] = (cc == code_buf[bnn]) ? norm_buf[bnn] : 0.f;
}

// ---------------------------------------------------------------------------
extern "C" void kernel_launch(void* const* d_in, const int* in_sizes, int n_in,
                              void* d_out, int out_size, void* d_ws, size_t ws_size,
                              hipStream_t stream)
{
    const float* pd_scores = (const float*)d_in[0];
    const float* pd_bboxes = (const float*)d_in[1];
    const float* anc       = (const float*)d_in[2];
    const int*   gt_labels = (const int*)d_in[3];
    const float* gt_bboxes = (const float*)d_in[4];
    const float* mask_gt   = (const float*)d_in[5];
    const float* gt_kkpts  = (const float*)d_in[6];
    const float* pd_kkpts  = (const float*)d_in[7];
    const float* sigma     = (const float*)d_in[8];
    const float* stride_t  = (const float*)d_in[9];
    const int*   reg_max_p = (const int*)d_in[10];

    const int na   = in_sizes[2] / 2;           // anc_points (na,2)
    const int nmax = in_sizes[3];               // gt_labels (bs,nmax,1)
    const int bs   = in_sizes[1] / (na * 4);    // pd_bboxes (bs,na,4)
    const int nc   = in_sizes[0] / (bs * na);   // pd_scores (bs,na,nc)
    const int nk   = in_sizes[8];               // sigma (nk,)

    const size_t bn  = (size_t)bs * na;
    const size_t bg  = (size_t)bs * nmax;
    const size_t bgn = bg * na;

    char* w = (char*)d_ws;
    float*    alignp  = (float*)w;    w += bgn * sizeof(float);
    float*    ovlp    = (float*)w;    w += bgn * sizeof(float);
    int*      cand    = (int*)w;      w += bg * TOPKN * sizeof(int);
    int*      fg_cnt  = (int*)w;      w += bn * sizeof(int);
    int*      tg_min  = (int*)w;      w += bn * sizeof(int);
    int*      tgi_buf = (int*)w;      w += bn * sizeof(int);
    unsigned* pa_bits = (unsigned*)w; w += bg * sizeof(unsigned);
    unsigned* po_bits = (unsigned*)w; w += bg * sizeof(unsigned);
    float*    norm_buf= (float*)w;    w += bn * sizeof(float);
    int*      code_buf= (int*)w;      w += bn * sizeof(int);

    hipMemsetAsync(fg_cnt, 0,    bn * sizeof(int), stream);
    hipMemsetAsync(tg_min, 0x7f, bn * sizeof(int), stream);
    hipMemsetAsync(pa_bits, 0,   bg * sizeof(unsigned), stream);
    hipMemsetAsync(po_bits, 0,   bg * sizeof(unsigned), stream);

    // K1: fused WMMA scores + pairwise metrics
    {
        const int tiles = (na + 15) / 16;
        dim3 grid((tiles + 3) / 4, (nmax + 15) / 16, bs);
        k1_pairwise<<<grid, 128, 0, stream>>>(pd_scores, pd_bboxes, anc, gt_labels,
                                              gt_bboxes, mask_gt, gt_kkpts, pd_kkpts,
                                              sigma, stride_t, reg_max_p,
                                              alignp, ovlp, na, nmax, nc, nk);
    }
    // K2: top-13 per gt row
    k2_topk<<<bs * nmax, 256, (size_t)na * sizeof(float), stream>>>(
        alignp, gt_bboxes, mask_gt, anc, stride_t, reg_max_p, cand, na, nmax);

    // K3a: scatter candidates
    {
        const int total = bs * nmax * TOPKN;
        k3a_scatter<<<(total + 255) / 256, 256, 0, stream>>>(cand, fg_cnt, tg_min,
                                                             na, nmax, total);
    }
    // K3b: resolve assignments + per-gt maxima
    {
        const int total = (int)bn;
        k3b_resolve<<<(total + 255) / 256, 256, 0, stream>>>(fg_cnt, tg_min, ovlp, alignp,
                                                             tgi_buf, pa_bits, po_bits,
                                                             na, nmax, total);
    }

    float* out   = (float*)d_out;
    float* o_lab = out;
    float* o_box = out + bn;
    float* o_sco = out + bn * 5;
    float* o_fg  = out + bn * 5 + bn * (size_t)nc;
    float* o_tgi = o_fg + bn;

    // K5: per-anchor outputs
    {
        const int total = (int)bn;
        k5_final<<<(total + 255) / 256, 256, 0, stream>>>(fg_cnt, tgi_buf, gt_labels,
                                                          gt_bboxes, alignp, pa_bits, po_bits,
                                                          o_lab, o_box, o_fg, o_tgi,
                                                          norm_buf, code_buf, na, nmax, total);
    }
    // K6: tscores
    {
        const long total = (long)bn * nc;
        k6_scores<<<(int)((total + 255) / 256), 256, 0, stream>>>(norm_buf, code_buf,
                                                                  o_sco, nc, total);
    }
}